// SVDNet_67740224193012
// MI455X (gfx1250) — compile-verified
//
#include <hip/hip_runtime.h>

// SVDNet on MI455X (gfx1250), compact-WY Householder chains.
//
//   y   = sigma ⊙ ( x + V1 * Tn1 * (VT_hhrs @ x) )         (VT chain, Tn = -T)
//   out =           y + U1ᵀ * Tn2 * (U_hhrs @ y)            (U chain)
//
// Traffic: 3 streaming reads of x + 1 write of out ≈ 512 MB (x stays resident
// in the 192MB L2 after pass 1).  All K=32 GEMM work runs on
// V_WMMA_F32_16X16X4_F32; tile staging uses GLOBAL_LOAD_ASYNC_TO_LDS_B128
// when the toolchain exposes it.

#define NDIM 4096
#define MDIM 8192
#define NH   32

typedef float v2f __attribute__((ext_vector_type(2)));
typedef float v8f __attribute__((ext_vector_type(8)));

#if defined(__gfx1250__) &&                                                    \
    __has_builtin(__builtin_amdgcn_global_load_async_to_lds_b128) &&           \
    __has_builtin(__builtin_amdgcn_s_wait_asynccnt)
#define USE_ASYNC_LDS 1
// Builtin expects: (v4i in AS1 (global/__device__), v4i in AS3 (LDS), imm, imm)
typedef int v4i __attribute__((vector_size(16)));
typedef __attribute__((address_space(1))) v4i as1_v4i;
typedef __attribute__((address_space(3))) v4i as3_v4i;
#else
#define USE_ASYNC_LDS 0
#endif

// Copy 16 bytes global -> LDS (async on CDNA5; tracked by ASYNCcnt).
static __device__ __forceinline__ void cp16_g2s(const float* g, float* l) {
#if USE_ASYNC_LDS
  __builtin_amdgcn_global_load_async_to_lds_b128((as1_v4i*)g, (as3_v4i*)l,
                                                 0, 0);
#else
  l[0] = g[0]; l[1] = g[1]; l[2] = g[2]; l[3] = g[3];
#endif
}

static __device__ __forceinline__ void cp_wait() {
#if USE_ASYNC_LDS
  __builtin_amdgcn_s_wait_asynccnt(0);
#endif
}

static __device__ __forceinline__ v8f wmma_f32_k4(v2f a, v2f b, v8f c) {
  // D = A(16x4) * B(4x16) + C(16x16), fp32
  return __builtin_amdgcn_wmma_f32_16x16x4_f32(false, a, false, b,
                                               (short)0, c, false, false);
}

// ---------------------------------------------------------------- Gram: G = H Hᵀ
__global__ __launch_bounds__(256) void svd_gram(const float* __restrict__ VT,
                                                const float* __restrict__ U,
                                                float* __restrict__ G) {
  const int bid = blockIdx.x;          // 2*NH*NH blocks
  const int which = bid >> 10;         // 0: VT, 1: U
  const int pair = bid & 1023;
  const int i = pair >> 5, j = pair & 31;
  const float* A = which ? U : VT;
  const float* hi = A + (size_t)i * NDIM;
  const float* hj = A + (size_t)j * NDIM;
  float s = 0.f;
  for (int n = threadIdx.x; n < NDIM; n += 256) s += hi[n] * hj[n];
  __shared__ float red[256];
  red[threadIdx.x] = s;
  __syncthreads();
  for (int st = 128; st > 0; st >>= 1) {
    if (threadIdx.x < st) red[threadIdx.x] += red[threadIdx.x + st];
    __syncthreads();
  }
  if (threadIdx.x == 0) G[which * NH * NH + i * NH + j] = red[0];
}

// --------------------------------------------- Tn = -T (lower triangular, WY)
// Tn[i][i] = -beta_i ; Tn[i][j<i] = -beta_i * sum_{l=j}^{i-1} G[i][l]*Tn[l][j]
__global__ void svd_buildT(const float* __restrict__ G, float* __restrict__ Tn) {
  const int tid = threadIdx.x;         // 64 threads: group g = matrix index
  const int g = tid >> 5, j = tid & 31;
  const float* Gg = G + g * NH * NH;
  __shared__ float T[2][NH][NH];
  for (int i = 0; i < NH; ++i) {
    const float beta = 2.0f / Gg[i * NH + i];
    float val = 0.f;
    if (j < i) {
      float s = 0.f;
      for (int l = j; l < i; ++l) s += Gg[i * NH + l] * T[g][l][j];
      val = -beta * s;
    } else if (j == i) {
      val = -beta;
    }
    T[g][i][j] = val;
    __syncthreads();
  }
  for (int i = 0; i < NH; ++i) Tn[g * NH * NH + i * NH + j] = T[g][i][j];
}

// ------------------------------------------------------------- Z = Tn @ W (32xM)
__global__ __launch_bounds__(256) void svd_applyT(const float* __restrict__ Tn,
                                                  const float* __restrict__ W,
                                                  float* __restrict__ Z) {
  __shared__ float Ts[NH][NH];
  const int tid = threadIdx.x;
  for (int idx = tid; idx < NH * NH; idx += 256) Ts[idx >> 5][idx & 31] = Tn[idx];
  __syncthreads();
  const int m = blockIdx.x * 256 + tid;
  float wv[NH];
#pragma unroll
  for (int k = 0; k < NH; ++k) wv[k] = W[(size_t)k * MDIM + m];
#pragma unroll
  for (int r = 0; r < NH; ++r) {
    float s = 0.f;
#pragma unroll
    for (int k = 0; k <= r; ++k) s += Ts[r][k] * wv[k];
    Z[(size_t)r * MDIM + m] = s;
  }
}

// ------------------------------------------------- pass1: W1 += VT @ x  (WMMA)
// grid (MDIM/64, NDIM/512); block 256 = 8 waves.
// Wave: colgroup cg = w%4 (16 cols), K-parity p = w>>2 (disjoint 4-row chunks).
__global__ __launch_bounds__(256) void svd_pass1(const float* __restrict__ x,
                                                 const float* __restrict__ VT,
                                                 float* __restrict__ W1) {
  __shared__ __align__(16) float vt_lds[NH][512 + 4];  // 2064B rows: 16B-aligned
  const int m0 = blockIdx.x * 64;
  const int n0 = blockIdx.y * 512;
  const int tid = threadIdx.x;
  for (int idx = tid; idx < NH * 128; idx += 256) {
    const int r = idx >> 7, c = (idx & 127) << 2;
    cp16_g2s(VT + (size_t)r * NDIM + n0 + c, &vt_lds[r][c]);
  }
  cp_wait();
  __syncthreads();
  const int w = tid >> 5, lane = tid & 31, lm = lane & 15, half = lane >> 4;
  const int cg = w & 3, p = w >> 2;
  const float* xcol = x + (size_t)m0 + 16 * cg + lm;
  v8f acc0 = {};
  v8f acc1 = {};
#pragma unroll 4
  for (int nn = p * 4; nn < 512; nn += 8) {
    const int k01 = nn + 2 * half;             // lanes16-31 hold K+2,K+3
    v2f a0, a1, b;
    a0.x = vt_lds[lm][k01];      a0.y = vt_lds[lm][k01 + 1];       // W1 rows 0-15
    a1.x = vt_lds[16 + lm][k01]; a1.y = vt_lds[16 + lm][k01 + 1];  // rows 16-31
    b.x = xcol[(size_t)(n0 + k01) * MDIM];
    b.y = xcol[(size_t)(n0 + k01 + 1) * MDIM];
    __builtin_prefetch(xcol + (size_t)(n0 + k01 + 32) * MDIM, 0, 1);
    acc0 = wmma_f32_k4(a0, b, acc0);
    acc1 = wmma_f32_k4(a1, b, acc1);
  }
  const int col = m0 + 16 * cg + lm;
#pragma unroll
  for (int v = 0; v < 8; ++v) {
    const int row = v + 8 * half;              // C/D layout: VGPR v -> M=v / v+8
    atomicAdd(&W1[(size_t)row * MDIM + col], acc0[v]);
    atomicAdd(&W1[(size_t)(16 + row) * MDIM + col], acc1[v]);
  }
}

// ----------------- pass2: y = sigma⊙(x + V1 Z1) fused with W2 += U @ y (WMMA)
__global__ __launch_bounds__(256) void svd_pass2(const float* __restrict__ x,
                                                 const float* __restrict__ VT,
                                                 const float* __restrict__ U,
                                                 const float* __restrict__ sigma,
                                                 const float* __restrict__ Z1,
                                                 float* __restrict__ W2) {
  __shared__ __align__(16) float vt_lds[NH][512 + 4];
  __shared__ __align__(16) float u_lds[NH][512 + 4];
  __shared__ __align__(16) float z_lds[NH][64 + 4];    // 272B rows: 16B-aligned
  __shared__ __align__(16) float sig_lds[512];
  __shared__ float y_lds[8][16][17];           // per-wave D->B relayout buffer
  const int m0 = blockIdx.x * 64;
  const int n0 = blockIdx.y * 512;
  const int tid = threadIdx.x;
  for (int idx = tid; idx < NH * 128; idx += 256) {
    const int r = idx >> 7, c = (idx & 127) << 2;
    cp16_g2s(VT + (size_t)r * NDIM + n0 + c, &vt_lds[r][c]);
    cp16_g2s(U + (size_t)r * NDIM + n0 + c, &u_lds[r][c]);
  }
  for (int idx = tid; idx < NH * 16; idx += 256) {
    const int r = idx >> 4, c = (idx & 15) << 2;
    cp16_g2s(Z1 + (size_t)r * MDIM + m0 + c, &z_lds[r][c]);
  }
  for (int idx = tid; idx < 128; idx += 256) {
    const int c = idx << 2;
    cp16_g2s(sigma + n0 + c, &sig_lds[c]);
  }
  cp_wait();
  __syncthreads();
  const int w = tid >> 5, lane = tid & 31, lm = lane & 15, half = lane >> 4;
  const int cg = w & 3, p = w >> 2;
  const int mcol = m0 + 16 * cg + lm;
  v8f w2a = {};
  v8f w2b = {};
  for (int nt = p; nt < 32; nt += 2) {         // 16 of 32 row-subtiles per wave
    const int nb = nt * 16;
    const float* xp = x + (size_t)(n0 + nb + 8 * half) * MDIM + mcol;
    v8f y;
#pragma unroll
    for (int v = 0; v < 8; ++v) y[v] = xp[(size_t)v * MDIM];   // C seeded with x
    __builtin_prefetch(xp + (size_t)32 * MDIM, 0, 1);
#pragma unroll
    for (int kk = 0; kk < NH; kk += 4) {       // y += V1 * Z1  (K=32)
      const int k01 = kk + 2 * half;
      v2f a, b;
      a.x = vt_lds[k01][nb + lm];      a.y = vt_lds[k01 + 1][nb + lm];  // V1=VTᵀ
      b.x = z_lds[k01][16 * cg + lm];  b.y = z_lds[k01 + 1][16 * cg + lm];
      y = wmma_f32_k4(a, b, y);
    }
#pragma unroll
    for (int v = 0; v < 8; ++v) y[v] *= sig_lds[nb + v + 8 * half];
#pragma unroll
    for (int v = 0; v < 8; ++v) y_lds[w][v + 8 * half][lm] = y[v];
#pragma unroll
    for (int kk = 0; kk < 16; kk += 4) {       // W2 += U * y  (K=16 rows of y)
      const int k01 = kk + 2 * half;
      v2f b, a0, a1;
      b.x = y_lds[w][k01][lm];         b.y = y_lds[w][k01 + 1][lm];
      a0.x = u_lds[lm][nb + k01];      a0.y = u_lds[lm][nb + k01 + 1];
      a1.x = u_lds[16 + lm][nb + k01]; a1.y = u_lds[16 + lm][nb + k01 + 1];
      w2a = wmma_f32_k4(a0, b, w2a);
      w2b = wmma_f32_k4(a1, b, w2b);
    }
  }
#pragma unroll
  for (int v = 0; v < 8; ++v) {
    const int row = v + 8 * half;
    atomicAdd(&W2[(size_t)row * MDIM + mcol], w2a[v]);
    atomicAdd(&W2[(size_t)(16 + row) * MDIM + mcol], w2b[v]);
  }
}

// --------- pass3: out = sigma⊙(x + V1 Z1) + U1ᵀ Z2  (recompute y, WMMA chain)
__global__ __launch_bounds__(256) void svd_pass3(const float* __restrict__ x,
                                                 const float* __restrict__ VT,
                                                 const float* __restrict__ U,
                                                 const float* __restrict__ sigma,
                                                 const float* __restrict__ Z1,
                                                 const float* __restrict__ Z2,
                                                 float* __restrict__ out) {
  __shared__ __align__(16) float vt_lds[NH][512 + 4];
  __shared__ __align__(16) float u_lds[NH][512 + 4];
  __shared__ __align__(16) float z1_lds[NH][64 + 4];
  __shared__ __align__(16) float z2_lds[NH][64 + 4];
  __shared__ __align__(16) float sig_lds[512];
  const int m0 = blockIdx.x * 64;
  const int n0 = blockIdx.y * 512;
  const int tid = threadIdx.x;
  for (int idx = tid; idx < NH * 128; idx += 256) {
    const int r = idx >> 7, c = (idx & 127) << 2;
    cp16_g2s(VT + (size_t)r * NDIM + n0 + c, &vt_lds[r][c]);
    cp16_g2s(U + (size_t)r * NDIM + n0 + c, &u_lds[r][c]);
  }
  for (int idx = tid; idx < NH * 16; idx += 256) {
    const int r = idx >> 4, c = (idx & 15) << 2;
    cp16_g2s(Z1 + (size_t)r * MDIM + m0 + c, &z1_lds[r][c]);
    cp16_g2s(Z2 + (size_t)r * MDIM + m0 + c, &z2_lds[r][c]);
  }
  for (int idx = tid; idx < 128; idx += 256) {
    const int c = idx << 2;
    cp16_g2s(sigma + n0 + c, &sig_lds[c]);
  }
  cp_wait();
  __syncthreads();
  const int w = tid >> 5, lane = tid & 31, lm = lane & 15, half = lane >> 4;
  const int cg = w & 3, p = w >> 2;
  const int mcol = m0 + 16 * cg + lm;
  for (int nt = p; nt < 32; nt += 2) {
    const int nb = nt * 16;
    const float* xp = x + (size_t)(n0 + nb + 8 * half) * MDIM + mcol;
    v8f y;
#pragma unroll
    for (int v = 0; v < 8; ++v) y[v] = xp[(size_t)v * MDIM];
    __builtin_prefetch(xp + (size_t)32 * MDIM, 0, 1);
#pragma unroll
    for (int kk = 0; kk < NH; kk += 4) {       // y += V1 * Z1
      const int k01 = kk + 2 * half;
      v2f a, b;
      a.x = vt_lds[k01][nb + lm];       a.y = vt_lds[k01 + 1][nb + lm];
      b.x = z1_lds[k01][16 * cg + lm];  b.y = z1_lds[k01 + 1][16 * cg + lm];
      y = wmma_f32_k4(a, b, y);
    }
#pragma unroll
    for (int v = 0; v < 8; ++v) y[v] *= sig_lds[nb + v + 8 * half];
#pragma unroll
    for (int kk = 0; kk < NH; kk += 4) {       // out = y + U1ᵀ * Z2
      const int k01 = kk + 2 * half;
      v2f a, b;
      a.x = u_lds[k01][nb + lm];        a.y = u_lds[k01 + 1][nb + lm];
      b.x = z2_lds[k01][16 * cg + lm];  b.y = z2_lds[k01 + 1][16 * cg + lm];
      y = wmma_f32_k4(a, b, y);
    }
    float* op = out + (size_t)(n0 + nb + 8 * half) * MDIM + mcol;
#pragma unroll
    for (int v = 0; v < 8; ++v)
      __builtin_nontemporal_store(y[v], op + (size_t)v * MDIM);  // TH=NT: keep
                                                                 // x hot in L2
  }
}

extern "C" void kernel_launch(void* const* d_in, const int* in_sizes, int n_in,
                              void* d_out, int out_size, void* d_ws, size_t ws_size,
                              hipStream_t stream) {
  (void)in_sizes; (void)n_in; (void)out_size; (void)ws_size;
  const float* x     = (const float*)d_in[0];   // [N, M]
  const float* sigma = (const float*)d_in[1];   // [N, 1]
  const float* U     = (const float*)d_in[2];   // [32, N]
  const float* VT    = (const float*)d_in[3];   // [32, N]
  float* out = (float*)d_out;

  float* ws = (float*)d_ws;
  float* G  = ws;                                  // 2*NH*NH
  float* Tn = G + 2 * NH * NH;                     // 2*NH*NH
  float* W1 = Tn + 2 * NH * NH;                    // NH*MDIM
  float* W2 = W1 + (size_t)NH * MDIM;              // NH*MDIM
  float* Z1 = W2 + (size_t)NH * MDIM;              // NH*MDIM
  float* Z2 = Z1 + (size_t)NH * MDIM;              // NH*MDIM  (~4.2 MB total)

  (void)hipMemsetAsync(W1, 0, 2 * (size_t)NH * MDIM * sizeof(float), stream);

  svd_gram<<<2 * NH * NH, 256, 0, stream>>>(VT, U, G);
  svd_buildT<<<1, 64, 0, stream>>>(G, Tn);

  dim3 grid(MDIM / 64, NDIM / 512);
  svd_pass1<<<grid, 256, 0, stream>>>(x, VT, W1);
  svd_applyT<<<MDIM / 256, 256, 0, stream>>>(Tn, W1, Z1);
  svd_pass2<<<grid, 256, 0, stream>>>(x, VT, U, sigma, Z1, W2);
  svd_applyT<<<MDIM / 256, 256, 0, stream>>>(Tn + NH * NH, W2, Z2);
  svd_pass3<<<grid, 256, 0, stream>>>(x, VT, U, sigma, Z1, Z2, out);
}